// MSAMG_13529146982473
// MI455X (gfx1250) — compile-verified
//
#include <hip/hip_runtime.h>

typedef float v2f __attribute__((ext_vector_type(2)));
typedef float v8f __attribute__((ext_vector_type(8)));
typedef _Float16 h2 __attribute__((ext_vector_type(2)));

#define HH 256
#define WW 256
#define HW 65536

static __device__ __forceinline__ v8f wmma4(v2f a, v2f b, v8f c) {
  return __builtin_amdgcn_wmma_f32_16x16x4_f32(false, a, false, b, (short)0, c, false, false);
}
static __device__ __forceinline__ float sigm(float v) {
  return 1.0f / (1.0f + __expf(-v));
}

// ---------------------------------------------------------------------------
// Kernel A: per (batch, group, 8-column strip): mhsa (WMMA MLPs + H-scan)
// writes t = mhsa output (f32) to ws, and CA-pool partials to rsum.
// ---------------------------------------------------------------------------
__global__ void __launch_bounds__(256) kA(
    const float* __restrict__ x,
    const float* __restrict__ w11, const float* __restrict__ w12,
    const float* __restrict__ w21, const float* __restrict__ w22,
    const float* __restrict__ b11, const float* __restrict__ b12,
    const float* __restrict__ b21, const float* __restrict__ b22,
    const float* __restrict__ spc_w1, const float* __restrict__ spc_b1,
    const float* __restrict__ spc_w2, const float* __restrict__ spc_b2,
    float* __restrict__ t, float* __restrict__ rsum)
{
  extern __shared__ char smem[];
  _Float16* zfh = (_Float16*)smem;                 // [256][32ch][8col][2] f16
  float* wl   = (float*)(smem + 256*32*8*2*2);     // 4 x [32][33]
  float* bl   = wl + 4*1056;                       // 4 x [32]
  float* scr0 = bl + 128;                          // 8 waves x [16][33]

  const int tid  = threadIdx.x;
  const int wave = tid >> 5, lane = tid & 31;
  const int m = lane & 15, hi = lane >> 4;
  const int w0 = blockIdx.x * 8;
  const int g  = blockIdx.y, b = blockIdx.z;
  const int cs = 24 * g;                           // START[g]

  for (int i = tid; i < 1024; i += 256) {
    const int o = i >> 5, c = i & 31;
    wl[0*1056 + o*33 + c] = w11[i];
    wl[1*1056 + o*33 + c] = w12[i];
    wl[2*1056 + o*33 + c] = w21[i];
    wl[3*1056 + o*33 + c] = w22[i];
  }
  if (tid < 128) {
    const float* bs = (tid < 32) ? b11 : (tid < 64) ? b12 : (tid < 96) ? b21 : b22;
    bl[tid] = bs[tid & 31];
  }
  __syncthreads();

  float* scr = scr0 + wave * 528;
  const float* xg = x + ((size_t)b*128 + cs)*HW + w0;

  // ---- phase 1: Z,F for all 2048 pixels (WMMA f32 16x16x4) -> LDS f16 ----
  for (int mt = wave; mt < 128; mt += 8) {
    const int pix = mt*16 + m;
    const int ph = pix >> 3, pc = pix & 7;
    const float* xp = xg + ph*WW + pc;
    float ax[8], ay[8];
#pragma unroll
    for (int kk = 0; kk < 8; ++kk) {
      const int ch = kk*4 + 2*hi;
      ax[kk] = xp[(size_t)ch*HW];
      ay[kk] = xp[(size_t)(ch + 1)*HW];
    }
#pragma unroll
    for (int ffn = 0; ffn < 2; ++ffn) {
      const float* W1 = wl + (ffn*2 + 0)*1056;
      const float* W2 = wl + (ffn*2 + 1)*1056;
      const float* B1 = bl + (ffn*2 + 0)*32;
      const float* B2 = bl + (ffn*2 + 1)*32;
      // layer 1: tanh(W1 x + b1)
#pragma unroll
      for (int nt = 0; nt < 2; ++nt) {
        const int n = nt*16 + m;
        const float bb = B1[n];
        v8f acc = {bb,bb,bb,bb,bb,bb,bb,bb};
#pragma unroll
        for (int kk = 0; kk < 8; ++kk) {
          const int k = kk*4 + 2*hi;
          v2f a; a.x = ax[kk]; a.y = ay[kk];
          v2f bv; bv.x = W1[n*33 + k]; bv.y = W1[n*33 + k + 1];
          acc = wmma4(a, bv, acc);
        }
#pragma unroll
        for (int r = 0; r < 8; ++r)
          scr[(hi*8 + r)*33 + n] = tanhf(acc[r]);
      }
      float hx[8], hy[8];
#pragma unroll
      for (int kk = 0; kk < 8; ++kk) {
        const int k = kk*4 + 2*hi;
        hx[kk] = scr[m*33 + k];
        hy[kk] = scr[m*33 + k + 1];
      }
      // layer 2: act(W2 h + b2) -> Z (tanh) or F (sigmoid)
#pragma unroll
      for (int nt = 0; nt < 2; ++nt) {
        const int n = nt*16 + m;
        const float bb = B2[n];
        v8f acc = {bb,bb,bb,bb,bb,bb,bb,bb};
#pragma unroll
        for (int kk = 0; kk < 8; ++kk) {
          const int k = kk*4 + 2*hi;
          v2f a; a.x = hx[kk]; a.y = hy[kk];
          v2f bv; bv.x = W2[n*33 + k]; bv.y = W2[n*33 + k + 1];
          acc = wmma4(a, bv, acc);
        }
#pragma unroll
        for (int r = 0; r < 8; ++r) {
          float v = acc[r];
          v = (ffn == 0) ? tanhf(v) : sigm(v);
          const int p2 = mt*16 + hi*8 + r;
          zfh[(((p2 >> 3)*32 + n)*8 + (p2 & 7))*2 + ffn] = (_Float16)v;
        }
      }
    }
  }
  __syncthreads();

  // ---- phase 2: 256 independent recurrences (c<16: reverse, c>=16: forward)
  {
    const int c = tid >> 3, col = tid & 7;
    const int fwd = (c >= 16);
    float y = 0.0f;
    float* tg = t + (((size_t)(b*5 + g)*32 + c)*HW) + w0 + col;
    h2* zf2 = (h2*)zfh;
    for (int step = 0; step < HH; ++step) {
      const int h = fwd ? step : (HH - 1 - step);
      const int idx = (h*32 + c)*8 + col;
      h2 v = zf2[idx];
      const float z = (float)v.x, f = (float)v.y;
      y = f*y + (1.0f - f)*z;
      tg[h*WW] = y;
      zfh[idx*2] = (_Float16)y;   // park y (f16) for phase 3 pooling
    }
  }
  __syncthreads();

  // ---- phase 3: block-partial of sum_pixels relu(spc_w1 y + spc_b1) -------
  float* red = scr0;  // 32 floats, scratch reuse
  {
    const float* w1g = spc_w1 + g*1024;
    for (int i = tid; i < 1024; i += 256)
      wl[(i >> 5)*33 + (i & 31)] = w1g[i];
    if (tid < 32) { bl[tid] = spc_b1[g*32 + tid]; red[tid] = 0.0f; }
  }
  __syncthreads();
  {
    float h1sum[32];
#pragma unroll
    for (int j = 0; j < 32; ++j) h1sum[j] = 0.0f;
    for (int i = 0; i < 8; ++i) {
      const int pix = tid + 256*i;
      const int ph = pix >> 3, pc = pix & 7;
      float yv[32];
#pragma unroll
      for (int c = 0; c < 32; ++c)
        yv[c] = (float)zfh[((ph*32 + c)*8 + pc)*2];
#pragma unroll
      for (int j = 0; j < 32; ++j) {
        float acc = bl[j];
#pragma unroll
        for (int c = 0; c < 32; ++c)
          acc += wl[j*33 + c] * yv[c];
        h1sum[j] += fmaxf(acc, 0.0f);
      }
    }
#pragma unroll
    for (int j = 0; j < 32; ++j)
      atomicAdd(&red[j], h1sum[j]);
  }
  __syncthreads();
  if (tid < 32) {
    const float* w2g = spc_w2 + g*1024 + tid*32;
    float v = 2048.0f * spc_b2[g*32 + tid];   // b2 folded per 2048-pixel block
#pragma unroll
    for (int j = 0; j < 32; ++j) v += w2g[j] * red[j];
    atomicAdd(&rsum[(b*5 + g)*32 + tid], v);
  }
}

// ---------------------------------------------------------------------------
// Kernel B: CA layer -> s; fold 0.1*midW*diag(s)*spcW2 and biases per (b,g)
// ---------------------------------------------------------------------------
__global__ void __launch_bounds__(32) kB(
    const float* __restrict__ rsum,
    const float* __restrict__ ca_w1, const float* __restrict__ ca_b1,
    const float* __restrict__ ca_w2, const float* __restrict__ ca_b2,
    const float* __restrict__ mid_w, const float* __restrict__ mid_b,
    const float* __restrict__ spc_w2, const float* __restrict__ spc_b2,
    float* __restrict__ w2c, float* __restrict__ mb)
{
  const int bg = blockIdx.x;        // b*5+g
  const int g = bg % 5;
  const int c = threadIdx.x;
  __shared__ float mean[32], sv[32];
  mean[c] = rsum[bg*32 + c] * (1.0f/65536.0f);
  __syncthreads();
  float v0 = ca_b1[g*2 + 0], v1 = ca_b1[g*2 + 1];
  for (int j = 0; j < 32; ++j) {
    v0 += ca_w1[(g*2 + 0)*32 + j] * mean[j];
    v1 += ca_w1[(g*2 + 1)*32 + j] * mean[j];
  }
  v0 = fmaxf(v0, 0.0f); v1 = fmaxf(v1, 0.0f);
  sv[c] = sigm(ca_b2[g*32 + c] + ca_w2[(g*32 + c)*2 + 0]*v0
                                + ca_w2[(g*32 + c)*2 + 1]*v1);
  __syncthreads();
  const int o = c;
  float acc[32];
#pragma unroll
  for (int j = 0; j < 32; ++j) acc[j] = 0.0f;
  float mbv = mid_b[g*32 + o];
  for (int cc = 0; cc < 32; ++cc) {
    const float coef = 0.1f * mid_w[g*1024 + o*32 + cc] * sv[cc];
    mbv += coef * spc_b2[g*32 + cc];
#pragma unroll
    for (int j = 0; j < 32; ++j)
      acc[j] += coef * spc_w2[g*1024 + cc*32 + j];
  }
#pragma unroll
  for (int j = 0; j < 32; ++j) w2c[bg*1024 + o*32 + j] = acc[j];
  mb[bg*32 + o] = mbv;
}

// ---------------------------------------------------------------------------
// Kernel C: per 16-pixel row tile: conv1+relu, fused mid, overlap-sum, tail
// ---------------------------------------------------------------------------
static __device__ __forceinline__ float cnt_inv(int c) {
  return (c >= 24 && c < 104 && ((c - 24) % 24) < 8) ? 0.5f : 1.0f;
}

__global__ void __launch_bounds__(256) kC(
    const float* __restrict__ t,
    const float* __restrict__ spc_w1, const float* __restrict__ spc_b1,
    const float* __restrict__ mid_w,
    const float* __restrict__ w2c, const float* __restrict__ mb,
    const float* __restrict__ tail_w, const float* __restrict__ tail_b,
    float* __restrict__ out)
{
  extern __shared__ char smem[];
  float* sW1  = (float*)smem;        // 5 x [32][33]
  float* sB1  = sW1 + 5*1056;        // 160
  float* sMW  = sB1 + 160;           // 5 x [32][33]
  float* sW2c = sMW + 5*1056;        // 5 x [32][33]
  float* sMB  = sW2c + 5*1056;       // 160
  float* sTW  = sMB + 160;           // [32][132] tail (cnt folded)
  float* sTB  = sTW + 4224;          // 32
  float* scr0 = sTB + 32;            // 8 x [16][33]
  float* ysc0 = scr0 + 8*528;        // 8 x [16][132]
  float* osc0 = ysc0 + 8*2112;       // 8 x [32][17]

  const int tid  = threadIdx.x;
  const int wave = tid >> 5, lane = tid & 31;
  const int m = lane & 15, hi = lane >> 4;
  const int mt0 = blockIdx.x * 32;
  const int b = mt0 >> 12;

  for (int i = tid; i < 5*1024; i += 256) {
    const int g = i >> 10, o = (i >> 5) & 31, c = i & 31;
    sW1[g*1056 + o*33 + c]  = spc_w1[i];
    sMW[g*1056 + o*33 + c]  = mid_w[i];
    sW2c[g*1056 + o*33 + c] = w2c[b*5120 + i];
  }
  for (int i = tid; i < 4096; i += 256) {
    const int o = i >> 7, c = i & 127;
    sTW[o*132 + c] = tail_w[i] * cnt_inv(c);
  }
  if (tid < 160) { sB1[tid] = spc_b1[tid]; sMB[tid] = mb[b*160 + tid]; }
  if (tid < 32) sTB[tid] = tail_b[tid];
  __syncthreads();

  float* scr = scr0 + wave*528;
  float* ysc = ysc0 + wave*2112;
  float* osc = osc0 + wave*544;

  for (int it = 0; it < 4; ++it) {
    const int id  = mt0 + wave*4 + it;
    const int row = (id >> 4) & 255;
    const int w0  = (id & 15) * 16;
    for (int i = lane; i < 2112; i += 32) ysc[i] = 0.0f;
    const float* tb = t + (size_t)b*5*32*HW + row*WW + w0;

    for (int g = 0; g < 5; ++g) {
      float tax[8], tay[8];
#pragma unroll
      for (int kk = 0; kk < 8; ++kk) {
        const int ch = kk*4 + 2*hi;
        tax[kk] = tb[(size_t)(g*32 + ch)*HW + m];
        tay[kk] = tb[(size_t)(g*32 + ch + 1)*HW + m];
      }
      // h1 = relu(spcW1 t + b1)
#pragma unroll
      for (int nt = 0; nt < 2; ++nt) {
        const int n = nt*16 + m;
        const float bb = sB1[g*32 + n];
        v8f acc = {bb,bb,bb,bb,bb,bb,bb,bb};
#pragma unroll
        for (int kk = 0; kk < 8; ++kk) {
          const int k = kk*4 + 2*hi;
          v2f a; a.x = tax[kk]; a.y = tay[kk];
          v2f bv; bv.x = sW1[g*1056 + n*33 + k]; bv.y = sW1[g*1056 + n*33 + k + 1];
          acc = wmma4(a, bv, acc);
        }
#pragma unroll
        for (int r = 0; r < 8; ++r)
          scr[(hi*8 + r)*33 + n] = fmaxf(acc[r], 0.0f);
      }
      float hx[8], hy[8];
#pragma unroll
      for (int kk = 0; kk < 8; ++kk) {
        const int k = kk*4 + 2*hi;
        hx[kk] = scr[m*33 + k];
        hy[kk] = scr[m*33 + k + 1];
      }
      // mid(resatt) = midW t + W2c h1 + mb   (CA scale + residual folded)
#pragma unroll
      for (int nt = 0; nt < 2; ++nt) {
        const int n = nt*16 + m;
        const float bb = sMB[g*32 + n];
        v8f acc = {bb,bb,bb,bb,bb,bb,bb,bb};
#pragma unroll
        for (int kk = 0; kk < 8; ++kk) {
          const int k = kk*4 + 2*hi;
          v2f a; a.x = tax[kk]; a.y = tay[kk];
          v2f bv; bv.x = sMW[g*1056 + n*33 + k]; bv.y = sMW[g*1056 + n*33 + k + 1];
          acc = wmma4(a, bv, acc);
        }
#pragma unroll
        for (int kk = 0; kk < 8; ++kk) {
          const int k = kk*4 + 2*hi;
          v2f a; a.x = hx[kk]; a.y = hy[kk];
          v2f bv; bv.x = sW2c[g*1056 + n*33 + k]; bv.y = sW2c[g*1056 + n*33 + k + 1];
          acc = wmma4(a, bv, acc);
        }
        const int cg = 24*g + n;
#pragma unroll
        for (int r = 0; r < 8; ++r)
          ysc[(hi*8 + r)*132 + cg] += acc[r];
      }
    }
    // tail: out = tailW' (y) + tail_b, K=128
#pragma unroll
    for (int nt = 0; nt < 2; ++nt) {
      const int n = nt*16 + m;
      const float bb = sTB[n];
      v8f acc = {bb,bb,bb,bb,bb,bb,bb,bb};
#pragma unroll
      for (int kk = 0; kk < 32; ++kk) {
        const int k = kk*4 + 2*hi;
        v2f a; a.x = ysc[m*132 + k]; a.y = ysc[m*132 + k + 1];
        v2f bv; bv.x = sTW[n*132 + k]; bv.y = sTW[n*132 + k + 1];
        acc = wmma4(a, bv, acc);
      }
#pragma unroll
      for (int r = 0; r < 8; ++r)
        osc[n*17 + hi*8 + r] = acc[r];
    }
    // coalesced store: lane == out channel, 64B contiguous per lane
    {
      float* og = out + ((size_t)b*32 + lane)*HW + row*WW + w0;
      float4* og4 = (float4*)og;
#pragma unroll
      for (int q = 0; q < 4; ++q) {
        float4 v;
        v.x = osc[lane*17 + q*4 + 0];
        v.y = osc[lane*17 + q*4 + 1];
        v.z = osc[lane*17 + q*4 + 2];
        v.w = osc[lane*17 + q*4 + 3];
        og4[q] = v;
      }
    }
  }
}

// ---------------------------------------------------------------------------
extern "C" void kernel_launch(void* const* d_in, const int* in_sizes, int n_in,
                              void* d_out, int out_size, void* d_ws, size_t ws_size,
                              hipStream_t stream) {
  (void)in_sizes; (void)n_in; (void)out_size; (void)ws_size;
  const float* x       = (const float*)d_in[0];
  const float* ffn1_w1 = (const float*)d_in[1];
  const float* ffn1_w2 = (const float*)d_in[2];
  const float* ffn2_w1 = (const float*)d_in[3];
  const float* ffn2_w2 = (const float*)d_in[4];
  const float* ffn1_b1 = (const float*)d_in[5];
  const float* ffn1_b2 = (const float*)d_in[6];
  const float* ffn2_b1 = (const float*)d_in[7];
  const float* ffn2_b2 = (const float*)d_in[8];
  const float* spc_w1  = (const float*)d_in[9];
  const float* spc_b1  = (const float*)d_in[10];
  const float* spc_w2  = (const float*)d_in[11];
  const float* spc_b2  = (const float*)d_in[12];
  const float* ca_w1   = (const float*)d_in[13];
  const float* ca_b1   = (const float*)d_in[14];
  const float* ca_w2   = (const float*)d_in[15];
  const float* ca_b2   = (const float*)d_in[16];
  const float* mid_w   = (const float*)d_in[17];
  const float* mid_b   = (const float*)d_in[18];
  const float* tail_w  = (const float*)d_in[19];
  const float* tail_b  = (const float*)d_in[20];
  float* out = (float*)d_out;

  float* t    = (float*)d_ws;                       // 2*5*32*65536 floats
  float* rsum = t + (size_t)2*5*32*HW;              // 320 floats
  float* w2cb = rsum + 320;                         // 10240 floats
  float* mbb  = w2cb + 10240;                       // 320 floats

  hipMemsetAsync(rsum, 0, 320*sizeof(float), stream);

  const size_t ldsA = (size_t)256*32*8*2*2 + (size_t)(4*1056 + 128 + 8*528)*sizeof(float);
  dim3 gA(32, 5, 2);
  kA<<<gA, 256, ldsA, stream>>>(x, ffn1_w1, ffn1_w2, ffn2_w1, ffn2_w2,
                                ffn1_b1, ffn1_b2, ffn2_b1, ffn2_b2,
                                spc_w1, spc_b1, spc_w2, spc_b2, t, rsum);

  kB<<<10, 32, 0, stream>>>(rsum, ca_w1, ca_b1, ca_w2, ca_b2,
                            mid_w, mid_b, spc_w2, spc_b2, w2cb, mbb);

  const size_t ldsC = (size_t)(5*1056*3 + 160*2 + 4224 + 32 + 8*528 + 8*2112 + 8*544)
                      * sizeof(float);
  kC<<<256, 256, ldsC, stream>>>(t, spc_w1, spc_b1, mid_w, w2cb, mbb,
                                 tail_w, tail_b, out);
}